// BahdanauAttention_65240553226467
// MI455X (gfx1250) — compile-verified
//
#include <hip/hip_runtime.h>
#include <cmath>

typedef float v2f __attribute__((ext_vector_type(2)));
typedef float v4f __attribute__((ext_vector_type(4)));
typedef float v8f __attribute__((ext_vector_type(8)));

// Problem shape (fixed by the harness/reference)
constexpr int Bsz = 4;
constexpr int Tq  = 256;
constexpr int Tk  = 512;
constexpr int Hd  = 512;
constexpr int W1S = 2 * Hd;   // w1 row stride (H, 2H) row-major

// ---- native tanh (gfx1250 V_TANH_F32 is a TRANS32 op) ----
__device__ __forceinline__ float fast_tanh(float x) {
#if __has_builtin(__builtin_amdgcn_tanhf)
    return __builtin_amdgcn_tanhf(x);
#elif __has_builtin(__builtin_amdgcn_tanh_f32)
    return __builtin_amdgcn_tanh_f32(x);
#else
    return tanhf(x);
#endif
}

// =====================================================================
// Projection GEMM: out[m, o] = sum_h In[m,h] * w1[o, hoff + h]  (+ bias[o])
// One wave per 16x16 output tile, V_WMMA_F32_16X16X4_F32, K stepped by 4.
// Exact f32 math (matches reference precision).
// =====================================================================
__global__ __launch_bounds__(256) void proj_wmma(const float* __restrict__ In,
                                                 const float* __restrict__ w1,
                                                 const float* __restrict__ bias, // may be null
                                                 float* __restrict__ out,
                                                 int M, int hoff) {
    const int NT   = Hd / 16;                 // 32 n-tiles
    const int tid  = threadIdx.x;
    const int wave = tid >> 5;
    const int lane = tid & 31;
    const int tile = blockIdx.x * 8 + wave;   // 8 waves per block, exact grids
    const int mt   = tile / NT;
    const int nt   = tile % NT;

    const int half = lane >> 4;               // 0: lanes 0-15, 1: lanes 16-31
    const int l15  = lane & 15;
    const int mrow = mt * 16 + l15;           // A row (M = lane&15 for both halves)
    const int ncol = nt * 16 + l15;           // B/D column

    const float* arow = In + (size_t)mrow * Hd;
    const float* brow = w1 + (size_t)ncol * W1S + hoff;

    v8f c = {0.f, 0.f, 0.f, 0.f, 0.f, 0.f, 0.f, 0.f};

#if __has_builtin(__builtin_amdgcn_wmma_f32_16x16x4_f32)
    for (int kk = 0; kk < Hd; kk += 4) {
        const int h0 = kk + half * 2;         // lanes 0-15: K=kk,kk+1 ; 16-31: K=kk+2,kk+3
        v2f a = *(const v2f*)(arow + h0);     // A[m][h0], A[m][h0+1]
        v2f b = *(const v2f*)(brow + h0);     // B[h0][n], B[h0+1][n]  (B[h][o] = w1[o,hoff+h])
        c = __builtin_amdgcn_wmma_f32_16x16x4_f32(false, a, false, b,
                                                  (short)0, c, false, false);
    }
#else
    // Scalar fallback (same output mapping)
    float acc[8] = {0.f, 0.f, 0.f, 0.f, 0.f, 0.f, 0.f, 0.f};
    for (int h = 0; h < Hd; ++h) {
        const float bvh = brow[h];
#pragma unroll
        for (int r = 0; r < 8; ++r)
            acc[r] += In[(size_t)(mt * 16 + half * 8 + r) * Hd + h] * bvh;
    }
#pragma unroll
    for (int r = 0; r < 8; ++r) c[r] = acc[r];
#endif

    const float bv = bias ? bias[ncol] : 0.0f;
#pragma unroll
    for (int r = 0; r < 8; ++r) {
        // 32-bit C/D 16x16 layout: VGPR r -> row (r + half*8), col = lane&15
        const int m = mt * 16 + half * 8 + r;
        out[(size_t)m * Hd + ncol] = c[r] + bv;
    }
}

// =====================================================================
// Fused tanh-score + mask + softmax. One workgroup per (b,q):
//   logit[k] = sum_h tanh(qp[b,q,h] + kp[b,k,h]) * w2[h] + b2
//   out[b,q,:] = softmax_k(mask ? logit : -1000)
// 512 threads = 16 wave32s; each wave covers 32 k rows; qp/w2 live in
// registers (each lane always touches the same 16 h-slots).
// =====================================================================
__global__ __launch_bounds__(512) void attn_fused(const float* __restrict__ qp,
                                                  const float* __restrict__ kp,
                                                  const int*   __restrict__ mask,
                                                  const float* __restrict__ w2,
                                                  const float* __restrict__ b2,
                                                  float* __restrict__ out) {
    __shared__ __align__(16) float logit_s[Tk];
    __shared__ float red_s[16];

    const int tid  = threadIdx.x;
    const int wave = tid >> 5;
    const int lane = tid & 31;
    const int bq   = blockIdx.x;          // 0 .. B*Tq-1
    const int b    = bq / Tq;
    const int q    = bq % Tq;

    // Per-lane register cache of qp row and w2 (16 values each, b128 loads)
    const float* qrow = qp + (size_t)bq * Hd;
    v4f qv[4], wv[4];
#pragma unroll
    for (int hh = 0; hh < 4; ++hh) {
        const int h = hh * 128 + lane * 4;
        qv[hh] = *(const v4f*)(qrow + h);
        wv[hh] = *(const v4f*)(w2 + h);
    }
    const float b2v = b2[0];
    const int*  mrow = mask + (size_t)bq * Tk;

    // 32 k rows per wave
    for (int i = 0; i < 32; ++i) {
        const int k = wave * 32 + i;
        const float* krow = kp + ((size_t)b * Tk + k) * Hd;
        float partial = 0.f;
#pragma unroll
        for (int hh = 0; hh < 4; ++hh) {
            const int h = hh * 128 + lane * 4;
            v4f kv = *(const v4f*)(krow + h);
            partial += fast_tanh(qv[hh].x + kv.x) * wv[hh].x;
            partial += fast_tanh(qv[hh].y + kv.y) * wv[hh].y;
            partial += fast_tanh(qv[hh].z + kv.z) * wv[hh].z;
            partial += fast_tanh(qv[hh].w + kv.w) * wv[hh].w;
        }
#pragma unroll
        for (int off = 16; off > 0; off >>= 1)
            partial += __shfl_down(partial, off, 32);
        if (lane == 0) {
            float lg = partial + b2v;
            if (mrow[k] == 0) lg = -1000.0f;
            logit_s[k] = lg;
        }
    }
    __syncthreads();

    // ---- softmax over k (512 values, 512 threads) ----
    const float v = logit_s[tid];

    float m = v;
#pragma unroll
    for (int off = 16; off > 0; off >>= 1)
        m = fmaxf(m, __shfl_down(m, off, 32));
    if (lane == 0) red_s[wave] = m;
    __syncthreads();
    float gmax = red_s[0];
#pragma unroll
    for (int w = 1; w < 16; ++w) gmax = fmaxf(gmax, red_s[w]);

    const float e = __expf(v - gmax);
    float s = e;
#pragma unroll
    for (int off = 16; off > 0; off >>= 1)
        s += __shfl_down(s, off, 32);
    __syncthreads();                      // all reads of red_s (max phase) done
    if (lane == 0) red_s[wave] = s;
    __syncthreads();
    float gsum = 0.f;
#pragma unroll
    for (int w = 0; w < 16; ++w) gsum += red_s[w];

    out[(size_t)bq * Tk + tid] = e / gsum;
}

// =====================================================================
extern "C" void kernel_launch(void* const* d_in, const int* in_sizes, int n_in,
                              void* d_out, int out_size, void* d_ws, size_t ws_size,
                              hipStream_t stream) {
    const float* query = (const float*)d_in[0];   // [B,Tq,H]
    const float* key   = (const float*)d_in[1];   // [B,Tk,H]
    const int*   mask  = (const int*)  d_in[2];   // [B,Tq,Tk]
    const float* w1    = (const float*)d_in[3];   // [H, 2H]
    const float* b1    = (const float*)d_in[4];   // [H]
    const float* w2    = (const float*)d_in[5];   // [A=1, H]
    const float* b2    = (const float*)d_in[6];   // [A=1]
    float* out = (float*)d_out;                   // [B,Tq,Tk,1]

    float* qp = (float*)d_ws;                     // [B*Tq, H] = 2 MB
    float* kp = qp + (size_t)Bsz * Tq * Hd;       // [B*Tk, H] = 4 MB

    // q_proj = query @ w1[:, :H]^T          (1024x512x512)
    {
        const int tiles = (Bsz * Tq / 16) * (Hd / 16);   // 2048
        proj_wmma<<<tiles / 8, 256, 0, stream>>>(query, w1, nullptr, qp, Bsz * Tq, 0);
    }
    // k_proj = key @ w1[:, H:]^T + b1       (2048x512x512)
    {
        const int tiles = (Bsz * Tk / 16) * (Hd / 16);   // 4096
        proj_wmma<<<tiles / 8, 256, 0, stream>>>(key, w1, b1, kp, Bsz * Tk, Hd);
    }
    // Fused tanh-score + mask + softmax, one WG per (b,q)
    attn_fused<<<Bsz * Tq, 512, 0, stream>>>(qp, kp, mask, w2, b2, out);
}